// PrimalDualNetwork_8675833938477
// MI455X (gfx1250) — compile-verified
//
#include <hip/hip_runtime.h>
#include <cstdint>

// ---------------------------------------------------------------------------
// Chambolle-Pock primal-dual TV denoising, 4096x4096 f32, 10 iterations.
// Memory-bound stencil: ~0.96 GB/iter -> ~0.41 ms total at 23.3 TB/s HBM.
// CDNA5 path used: GLOBAL_LOAD_ASYNC_TO_LDS_* staging + s_wait_asynccnt,
// with direct float4 register loads issued *between* the async issue and the
// asynccnt wait so HBM latency of both streams overlaps.
// ---------------------------------------------------------------------------

namespace {

constexpr int    NDIM   = 4096;
constexpr size_t NN     = (size_t)NDIM * NDIM;
constexpr float  SIGMA  = 1.0f / (7.0f * 0.01f);
constexpr float  TAU    = 0.01f;
constexpr float  LAMROF = 4.0f;
constexpr float  THETA  = 0.5f;
constexpr float  INVDEN = 1.0f / (1.0f + LAMROF * TAU);

constexpr int TW = 64;   // tile width  (pixels)
constexpr int TH = 16;   // tile height (pixels)
constexpr int LS = 68;   // LDS row stride in floats (68*4 = 272 B, 16B aligned)

// ---- async global -> LDS copy helpers (gfx1250) ---------------------------
#if defined(__AMDGCN__) &&                                                  \
    __has_builtin(__builtin_amdgcn_global_load_async_to_lds_b128) &&        \
    __has_builtin(__builtin_amdgcn_global_load_async_to_lds_b32) &&         \
    __has_builtin(__builtin_amdgcn_s_wait_asynccnt)
#define PD_USE_ASYNC 1
#else
#define PD_USE_ASYNC 0
#endif

#if PD_USE_ASYNC
typedef int v4i __attribute__((ext_vector_type(4)));
typedef __attribute__((address_space(1))) v4i* Gv4P;   // global v4i32*
typedef __attribute__((address_space(1))) int* GiP;    // global i32*
typedef __attribute__((address_space(3))) v4i* Lv4P;   // lds v4i32*
typedef __attribute__((address_space(3))) int* LiP;    // lds i32*
#endif

__device__ __forceinline__ void async_copy16(const float* g, float* l) {
#if PD_USE_ASYNC
  __builtin_amdgcn_global_load_async_to_lds_b128((Gv4P)g, (Lv4P)l, 0, 0);
#else
  *(float4*)l = *(const float4*)g;
#endif
}

__device__ __forceinline__ void async_copy4(const float* g, float* l) {
#if PD_USE_ASYNC
  __builtin_amdgcn_global_load_async_to_lds_b32((GiP)g, (LiP)l, 0, 0);
#else
  *l = *g;
#endif
}

__device__ __forceinline__ void async_join() {
#if PD_USE_ASYNC
  __builtin_amdgcn_s_wait_asynccnt(0);  // all async LDS writes landed
#endif
  // (fallback path: plain ds stores are covered by __syncthreads)
}

__device__ __forceinline__ int imin(int a, int b) { return a < b ? a : b; }
__device__ __forceinline__ float clip1(float v) {
  return fminf(1.0f, fmaxf(-1.0f, v));
}

// ---------------------------------------------------------------------------
// Dual update:  y = clip(y + SIGMA * w * fwd_grad(x_tilde), -1, 1)
//   gh(r,c) = xt(r,c+1)-xt(r,c)  (0 at c=N-1)   [halo clamped -> diff==0]
//   gv(r,c) = xt(r+1,c)-xt(r,c)  (0 at r=N-1)
// FIRST: y_in treated as all-ones (skip the 128 MB y read on iteration 0).
// ---------------------------------------------------------------------------
template <bool FIRST>
__global__ void __launch_bounds__(256)
pd_dual_kernel(const float* __restrict__ xt,
               const float* __restrict__ w,
               float* __restrict__ y) {
  __shared__ float xs[(TH + 1) * LS];

  const int c0 = blockIdx.x * TW;
  const int r0 = blockIdx.y * TH;
  const int t  = threadIdx.x;

  // --- issue async staging: (TH+1) x (TW+1) x_tilde tile, clamped halos ----
  for (int idx = t; idx < (TH + 1) * 16; idx += 256) {   // 272 x b128
    const int rr = idx >> 4, q = idx & 15;
    const int gr = imin(r0 + rr, NDIM - 1);
    async_copy16(&xt[(size_t)gr * NDIM + c0 + q * 4], &xs[rr * LS + q * 4]);
  }
  if (t <= TH) {                                          // 17 x b32 halo col
    const int gr = imin(r0 + t, NDIM - 1);
    const int gc = imin(c0 + TW, NDIM - 1);
    async_copy4(&xt[(size_t)gr * NDIM + gc], &xs[t * LS + TW]);
  }

  // --- issue independent direct loads while the async DMA is in flight -----
  const int    tx = t & 15, ty = t >> 4;
  const int    c  = c0 + tx * 4;
  const size_t g  = (size_t)(r0 + ty) * NDIM + c;

  const float4 w0 = *(const float4*)&w[g];
  const float4 w1 = *(const float4*)&w[NN + g];
  float4 y0, y1;
  if (FIRST) {
    y0 = make_float4(1.f, 1.f, 1.f, 1.f);
    y1 = make_float4(1.f, 1.f, 1.f, 1.f);
  } else {
    y0 = *(const float4*)&y[g];
    y1 = *(const float4*)&y[NN + g];
  }

  // --- join ---------------------------------------------------------------
  async_join();
  __syncthreads();

  const float* xr0 = &xs[ty * LS + tx * 4];        // this row (5 values used)
  const float* xr1 = &xs[(ty + 1) * LS + tx * 4];  // row below

  float o0[4], o1[4];
  const float yi0[4] = {y0.x, y0.y, y0.z, y0.w};
  const float yi1[4] = {y1.x, y1.y, y1.z, y1.w};
  const float wi0[4] = {w0.x, w0.y, w0.z, w0.w};
  const float wi1[4] = {w1.x, w1.y, w1.z, w1.w};
#pragma unroll
  for (int k = 0; k < 4; ++k) {
    const float xc = xr0[k];
    const float gh = xr0[k + 1] - xc;  // slot TW holds the clamped halo col
    const float gv = xr1[k] - xc;
    o0[k] = clip1(fmaf(SIGMA * wi0[k], gh, yi0[k]));
    o1[k] = clip1(fmaf(SIGMA * wi1[k], gv, yi1[k]));
  }
  *(float4*)&y[g]      = make_float4(o0[0], o0[1], o0[2], o0[3]);
  *(float4*)&y[NN + g] = make_float4(o1[0], o1[1], o1[2], o1[3]);
}

// ---------------------------------------------------------------------------
// Primal update:
//   a = w0*y0 ; b = w1*y1
//   d_h(r,c) = (c<N-1 ? a(r,c) : 0) - (c>0 ? a(r,c-1) : 0)
//   d_v(r,c) = (r<N-1 ? b(r,c) : 0) - (r>0 ? b(r-1,c) : 0)
//   x_new  = (x + TAU*(d_h+d_v) + LAM*TAU*img) / (1+LAM*TAU)
//   x_tilde = x_new + THETA*(x_new - x)
// FIRST: x_in == img (one read serves both terms).
// NOTE: x_in / x_out alias in-place for iterations >= 1 -> no __restrict__.
// ---------------------------------------------------------------------------
template <bool FIRST>
__global__ void __launch_bounds__(256)
pd_primal_kernel(const float* x_in,
                 const float* __restrict__ y,
                 const float* __restrict__ w,
                 const float* __restrict__ img,
                 float* x_out,
                 float* __restrict__ xt_out) {
  // y0/w0: rows r0..r0+TH-1, cols c0-1..c0+TW-1 (halo col in slot 3, data 4..67)
  // y1/w1: rows r0-1..r0+TH-1 (halo row = index 0), cols c0..c0+TW-1 (slots 0..63)
  __shared__ float y0s[TH * LS];
  __shared__ float w0s[TH * LS];
  __shared__ float y1s[(TH + 1) * LS];
  __shared__ float w1s[(TH + 1) * LS];

  const int c0 = blockIdx.x * TW;
  const int r0 = blockIdx.y * TH;
  const int t  = threadIdx.x;

  const float* y0p = y;
  const float* y1p = y + NN;
  const float* w0p = w;
  const float* w1p = w + NN;

  // --- issue async staging -------------------------------------------------
  {  // y0s / w0s main body: 256 quads each, one per thread
    const int rr = t >> 4, q = t & 15;
    const size_t gq = (size_t)(r0 + rr) * NDIM + c0 + q * 4;
    async_copy16(&y0p[gq], &y0s[rr * LS + 4 + q * 4]);
    async_copy16(&w0p[gq], &w0s[rr * LS + 4 + q * 4]);
  }
  if (t < 32) {  // left halo column (zero at domain edge)
    const int rr = t & 15;
    float*       dst = (t < 16) ? &y0s[rr * LS + 3] : &w0s[rr * LS + 3];
    const float* src = (t < 16) ? y0p : w0p;
    if (c0 == 0) *dst = 0.0f;
    else async_copy4(&src[(size_t)(r0 + rr) * NDIM + (c0 - 1)], dst);
  }
  for (int idx = t; idx < (TH + 1) * 16; idx += 256) {  // y1s / w1s (17 rows)
    const int rr = idx >> 4, q = idx & 15;
    const int gr = r0 + rr - 1;
    if (gr < 0) {  // zero-fill top halo at domain edge
      const float4 z = make_float4(0.f, 0.f, 0.f, 0.f);
      *(float4*)&y1s[rr * LS + q * 4] = z;
      *(float4*)&w1s[rr * LS + q * 4] = z;
    } else {
      const size_t gq = (size_t)gr * NDIM + c0 + q * 4;
      async_copy16(&y1p[gq], &y1s[rr * LS + q * 4]);
      async_copy16(&w1p[gq], &w1s[rr * LS + q * 4]);
    }
  }

  // --- issue independent direct loads while the async DMA is in flight -----
  const int    tx = t & 15, ty = t >> 4;
  const int    c  = c0 + tx * 4;
  const int    r  = r0 + ty;
  const size_t g  = (size_t)r * NDIM + c;

  const float4 xv = *(const float4*)&x_in[g];
  const float4 iv = FIRST ? xv : *(const float4*)&img[g];

  // --- join ---------------------------------------------------------------
  async_join();
  __syncthreads();

  const float xi[4] = {xv.x, xv.y, xv.z, xv.w};
  const float fi[4] = {iv.x, iv.y, iv.z, iv.w};

  const float* a0 = &y0s[ty * LS + 3 + tx * 4];  // a offsets: [0]=c-1 .. [4]=c+3
  const float* aw = &w0s[ty * LS + 3 + tx * 4];
  const float* bU = &y1s[ty * LS + tx * 4];      // row r-1 (halo row when ty==0)
  const float* bUw= &w1s[ty * LS + tx * 4];
  const float* bC = &y1s[(ty + 1) * LS + tx * 4];// row r
  const float* bCw= &w1s[(ty + 1) * LS + tx * 4];

  const float rowMask = (r == NDIM - 1) ? 0.0f : 1.0f;

  float ox[4], oxt[4];
  float am1 = a0[0] * aw[0];  // a(r, c-1)  (zero at c==0)
#pragma unroll
  for (int k = 0; k < 4; ++k) {
    const float ak  = a0[k + 1] * aw[k + 1];
    const float dh  = (((c + k) == NDIM - 1) ? 0.0f : ak) - am1;
    am1 = ak;
    const float dv  = rowMask * (bC[k] * bCw[k]) - bU[k] * bUw[k];
    const float xn  = (fmaf(TAU, dh + dv, xi[k]) + (LAMROF * TAU) * fi[k]) * INVDEN;
    ox[k]  = xn;
    oxt[k] = fmaf(THETA, xn - xi[k], xn);  // (1+THETA)*xn - THETA*x_old
  }
  *(float4*)&x_out[g]  = make_float4(ox[0], ox[1], ox[2], ox[3]);
  *(float4*)&xt_out[g] = make_float4(oxt[0], oxt[1], oxt[2], oxt[3]);
}

}  // namespace

// ---------------------------------------------------------------------------
// Host driver: 10 iterations, 2 fused kernels each.
//   x_tilde lives in d_out (its final write IS the output).
//   d_ws: [x (16M floats)] [y0 (16M)] [y1 (16M)]  => 192 MB scratch.
// ---------------------------------------------------------------------------
extern "C" void kernel_launch(void* const* d_in, const int* in_sizes, int n_in,
                              void* d_out, int out_size, void* d_ws, size_t ws_size,
                              hipStream_t stream) {
  (void)in_sizes; (void)n_in; (void)out_size; (void)ws_size;

  const float* img = (const float*)d_in[0];  // [1,4096,4096]
  const float* w   = (const float*)d_in[1];  // [2,4096,4096]

  float* xt   = (float*)d_out;               // x_tilde (final answer)
  float* xbuf = (float*)d_ws;                // x
  float* ybuf = xbuf + NN;                   // y (2 planes, contiguous)

  const dim3 grid(NDIM / TW, NDIM / TH);     // 64 x 256 tiles
  const dim3 block(256);

  // Iteration 0: x_tilde == img, y == 1, x == img (all folded in, no init pass)
  pd_dual_kernel<true><<<grid, block, 0, stream>>>(img, w, ybuf);
  pd_primal_kernel<true><<<grid, block, 0, stream>>>(img, ybuf, w, img, xbuf, xt);

  for (int it = 1; it < 10; ++it) {
    pd_dual_kernel<false><<<grid, block, 0, stream>>>(xt, w, ybuf);
    pd_primal_kernel<false><<<grid, block, 0, stream>>>(xbuf, ybuf, w, img, xbuf, xt);
  }
}